// ToggleGNN_90855738180233
// MI455X (gfx1250) — compile-verified
//
#include <hip/hip_runtime.h>

// ---------------------------------------------------------------------------
// GraphSAGE (2x SAGEConv mean-aggr + FC head) for MI455X / gfx1250.
//
// Roofline: edge gather/scatter (~0.9 GB through L2-resident buffers)
// dominates; dense GEMMs (~2.5 GFLOP) run on V_WMMA_F32_16X16X4_F32.
// All node buffers fit in the 192MB global L2, so random gathers and
// global_atomic_add_f32 scatters stay on-chip.
//
// GEMM strategy: K padded to a multiple of 4 (35 -> 36) and weights
// pre-transposed to [64][Kpad], so every WMMA fragment is a contiguous
// 8-byte load with zero bounds checks. A-tiles (shared by all 4 waves of
// a block) and both weight matrices are staged in LDS; inner loop is pure
// ds_load_b64 + v_wmma_f32_16x16x4_f32, fully unrolled.
// ---------------------------------------------------------------------------

typedef __attribute__((ext_vector_type(2))) float v2f;
typedef __attribute__((ext_vector_type(8))) float v8f;

#define HID 64
#define LP 68   // LDS row stride (floats): LP % 64-bank pattern => conflict-free

// ---- zero fill (grid-stride) ----------------------------------------------
__global__ void zero_f32(float* __restrict__ p, long n) {
  long i = (long)blockIdx.x * blockDim.x + threadIdx.x;
  long stride = (long)gridDim.x * blockDim.x;
  for (; i < n; i += stride) p[i] = 0.0f;
}

// ---- pad-copy: xp[N][KP] = x[N][K] zero-padded ----------------------------
__global__ void pad_copy_kernel(const float* __restrict__ x,
                                float* __restrict__ xp,
                                int N, int K, int KP) {
  long i = (long)blockIdx.x * blockDim.x + threadIdx.x;
  long total = (long)N * KP;
  if (i >= total) return;
  int r = (int)(i / KP);
  int c = (int)(i % KP);
  xp[i] = (c < K) ? x[(long)r * K + c] : 0.0f;
}

// ---- weight pack: Wt[64][KP] = transpose(W[K][64]) zero-padded ------------
__global__ void pack_wt_kernel(const float* __restrict__ W,
                               float* __restrict__ Wt, int K, int KP) {
  int i = blockIdx.x * blockDim.x + threadIdx.x;
  if (i >= HID * KP) return;
  int col = i / KP;                 // output channel
  int k   = i % KP;
  Wt[i] = (k < K) ? W[k * HID + col] : 0.0f;
}

// ---- in-degree via f32 atomics --------------------------------------------
__global__ void degree_kernel(const int* __restrict__ dst, int E,
                              float* __restrict__ deg) {
  int e = blockIdx.x * blockDim.x + threadIdx.x;
  if (e < E) atomicAdd(&deg[dst[e]], 1.0f);
}

// ---- edge scatter-add: agg[dst] += feat[src]; F % 4 == 0, no branches -----
__global__ void scatter_add_kernel(const float* __restrict__ feat,
                                   const int* __restrict__ src,
                                   const int* __restrict__ dst,
                                   int E, int F, int chunks,
                                   float* __restrict__ agg) {
  long t = (long)blockIdx.x * blockDim.x + threadIdx.x;
  long total = (long)E * chunks;
  if (t >= total) return;
  int e  = (int)(t / chunks);
  int c0 = (int)(t % chunks) * 4;
  const float* fs = feat + (long)src[e] * F + c0;
  float*       ad = agg  + (long)dst[e] * F + c0;
#pragma unroll
  for (int j = 0; j < 4; ++j) atomicAdd(&ad[j], fs[j]);
}

// ---- mean = agg / max(deg, 1) (in place, padded cols stay zero) -----------
__global__ void mean_kernel(float* __restrict__ agg,
                            const float* __restrict__ deg, int N, int F) {
  int i = blockIdx.x * blockDim.x + threadIdx.x;
  if (i >= N) return;
  float inv = 1.0f / fmaxf(deg[i], 1.0f);
  float* row = agg + (long)i * F;
  for (int c = 0; c < F; ++c) row[c] *= inv;
}

// ---- fused SAGE GEMM: out = relu(Am @ Wl^T' + As @ Wr^T' + bias) ----------
// Am, As: [N][KPAD] row-major (zero-padded K). Wlt, Wrt: [64][KPAD]
// (pre-transposed, zero-padded). One wave per 16x16 output tile, 4 waves
// per block covering the 64 output channels.
template <int KPAD>
__global__ __launch_bounds__(128)
void sage_gemm_kernel(const float* __restrict__ Am,
                      const float* __restrict__ As,
                      const float* __restrict__ Wlt,
                      const float* __restrict__ Wrt,
                      const float* __restrict__ bias,
                      float* __restrict__ out, int Nn) {
  __shared__ float sAm[16 * LP];
  __shared__ float sAs[16 * LP];
  __shared__ float sWl[HID * LP];
  __shared__ float sWr[HID * LP];

  const int tid = threadIdx.x;
  const int m0  = blockIdx.x * 16;

  // stage A tiles (coalesced; shared by all 4 waves)
  for (int idx = tid; idx < 16 * KPAD; idx += 128) {
    int r = idx / KPAD, c = idx % KPAD;
    int row = m0 + r;
    bool ok = (row < Nn);
    long g = (long)row * KPAD + c;
    sAm[r * LP + c] = ok ? Am[g] : 0.0f;
    sAs[r * LP + c] = ok ? As[g] : 0.0f;
  }
  // stage both weight matrices
  for (int idx = tid; idx < HID * KPAD; idx += 128) {
    int r = idx / KPAD, c = idx % KPAD;
    sWl[r * LP + c] = Wlt[idx];
    sWr[r * LP + c] = Wrt[idx];
  }
  __syncthreads();

  const int wave = tid >> 5;
  const int lane = tid & 31;
  const int hi   = lane >> 4;        // 0/1 -> K split per A/B f32 layout
  const int lo   = lane & 15;
  const int bcol = wave * 16 + lo;

  const float* pAm = &sAm[lo * LP + 2 * hi];
  const float* pAs = &sAs[lo * LP + 2 * hi];
  const float* pWl = &sWl[bcol * LP + 2 * hi];
  const float* pWr = &sWr[bcol * LP + 2 * hi];

  v8f acc = {};
#pragma unroll
  for (int k = 0; k < KPAD; k += 4) {
    v2f a, b;
    // neighbor-mean path
    a.x = pAm[k]; a.y = pAm[k + 1];
    b.x = pWl[k]; b.y = pWl[k + 1];
    acc = __builtin_amdgcn_wmma_f32_16x16x4_f32(
        false, a, false, b, (short)0, acc, false, false);
    // self path
    a.x = pAs[k]; a.y = pAs[k + 1];
    b.x = pWr[k]; b.y = pWr[k + 1];
    acc = __builtin_amdgcn_wmma_f32_16x16x4_f32(
        false, a, false, b, (short)0, acc, false, false);
  }

  const float bv = bias[bcol];
#pragma unroll
  for (int r = 0; r < 8; ++r) {
    int row = m0 + r + 8 * hi;
    if (row < Nn)
      out[(long)row * HID + bcol] = fmaxf(acc[r] + bv, 0.0f);  // fused ReLU
  }
}

// ---- FC head: out[i] = h2[i,:] . Wfc + bfc --------------------------------
__global__ void fc_kernel(const float* __restrict__ h,
                          const float* __restrict__ Wfc,
                          const float* __restrict__ bfc,
                          float* __restrict__ out, int N) {
  int i = blockIdx.x * blockDim.x + threadIdx.x;
  if (i >= N) return;
  const float4* row = (const float4*)(h + (long)i * HID);
  const float4* w   = (const float4*)Wfc;
  float s = bfc[0];
#pragma unroll
  for (int c = 0; c < HID / 4; ++c) {
    float4 r4 = row[c], w4 = w[c];
    s = fmaf(r4.x, w4.x, s);
    s = fmaf(r4.y, w4.y, s);
    s = fmaf(r4.z, w4.z, s);
    s = fmaf(r4.w, w4.w, s);
  }
  out[i] = s;
}

// ---------------------------------------------------------------------------
extern "C" void kernel_launch(void* const* d_in, const int* in_sizes, int n_in,
                              void* d_out, int out_size, void* d_ws, size_t ws_size,
                              hipStream_t stream) {
  const float* x   = (const float*)d_in[0];   // [N, 35]
  const int*   ei  = (const int*)d_in[1];     // [2, E]
  const float* W1l = (const float*)d_in[2];   // [35, 64]
  const float* b1  = (const float*)d_in[3];   // [64]
  const float* W1r = (const float*)d_in[4];   // [35, 64]
  const float* W2l = (const float*)d_in[5];   // [64, 64]
  const float* b2  = (const float*)d_in[6];   // [64]
  const float* W2r = (const float*)d_in[7];   // [64, 64]
  const float* Wfc = (const float*)d_in[8];   // [64, 1]
  const float* bfc = (const float*)d_in[9];   // [1]

  const int IN_CH = 35;
  const int KP1   = 36;                        // padded K for layer 1
  const int N = in_sizes[0] / IN_CH;           // 100000
  const int E = in_sizes[1] / 2;               // 1600000
  const int* src = ei;
  const int* dst = ei + E;

  // workspace carve: deg | agg(N*64) | h1(N*64) | h2(N*64) | xp(N*36) | Wt's
  float* deg  = (float*)d_ws;
  float* agg  = deg  + N;
  float* h1   = agg  + (long)N * HID;
  float* h2   = h1   + (long)N * HID;
  float* xp   = h2   + (long)N * HID;
  float* W1lt = xp   + (long)N * KP1;
  float* W1rt = W1lt + HID * KP1;
  float* W2lt = W1rt + HID * KP1;
  float* W2rt = W2lt + HID * HID;

  const int TB = 256;

  // ---- prep: padded x, transposed+padded weights, degree, zero agg ----
  zero_f32<<<8192, TB, 0, stream>>>(deg, (long)N * (1 + KP1)); // deg + agg(L1)
  pad_copy_kernel<<<(int)(((long)N * KP1 + TB - 1) / TB), TB, 0, stream>>>(
      x, xp, N, IN_CH, KP1);
  pack_wt_kernel<<<(HID * KP1 + TB - 1) / TB, TB, 0, stream>>>(W1l, W1lt, IN_CH, KP1);
  pack_wt_kernel<<<(HID * KP1 + TB - 1) / TB, TB, 0, stream>>>(W1r, W1rt, IN_CH, KP1);
  pack_wt_kernel<<<(HID * HID + TB - 1) / TB, TB, 0, stream>>>(W2l, W2lt, HID, HID);
  pack_wt_kernel<<<(HID * HID + TB - 1) / TB, TB, 0, stream>>>(W2r, W2rt, HID, HID);
  degree_kernel<<<(E + TB - 1) / TB, TB, 0, stream>>>(dst, E, deg);

  // ---- layer 1: agg = sum_{j->i} xp_j ; mean ; WMMA GEMM + ReLU ----
  {
    const int F = KP1, CH = KP1 / 4;            // 9 chunks, branch-free
    long tot = (long)E * CH;
    scatter_add_kernel<<<(int)((tot + TB - 1) / TB), TB, 0, stream>>>(
        xp, src, dst, E, F, CH, agg);
    mean_kernel<<<(N + TB - 1) / TB, TB, 0, stream>>>(agg, deg, N, F);
    sage_gemm_kernel<KP1><<<(N + 15) / 16, 128, 0, stream>>>(
        agg, xp, W1lt, W1rt, b1, h1, N);
  }

  // ---- layer 2: agg = sum_{j->i} h1_j ; mean ; WMMA GEMM + ReLU ----
  {
    const int F = HID, CH = HID / 4;            // 16 chunks, branch-free
    zero_f32<<<8192, TB, 0, stream>>>(agg, (long)N * HID);
    long tot = (long)E * CH;
    scatter_add_kernel<<<(int)((tot + TB - 1) / TB), TB, 0, stream>>>(
        h1, src, dst, E, F, CH, agg);
    mean_kernel<<<(N + TB - 1) / TB, TB, 0, stream>>>(agg, deg, N, F);
    sage_gemm_kernel<HID><<<(N + 15) / 16, 128, 0, stream>>>(
        agg, h1, W2lt, W2rt, b2, h2, N);
  }

  // ---- FC head ----
  fc_kernel<<<(N + TB - 1) / TB, TB, 0, stream>>>(h2, Wfc, bfc,
                                                  (float*)d_out, N);
}